// ClusterModel_88974542504688
// MI455X (gfx1250) — compile-verified
//
#include <hip/hip_runtime.h>

typedef float v2f __attribute__((ext_vector_type(2)));
typedef float v4f __attribute__((ext_vector_type(4)));
typedef float v8f __attribute__((ext_vector_type(8)));

#define G  8192
#define S  64
#define D  128
#define H  256
#define KC 8
#define EPS 0.01f

__device__ __forceinline__ v8f wmma_f32_16x16x4(v2f a, v2f b, v8f c) {
  // 8 args: (neg_a, A, neg_b, B, c_mod, C, reuse_a, reuse_b)
  return __builtin_amdgcn_wmma_f32_16x16x4_f32(false, a, false, b, (short)0, c,
                                               false, false);
}

__device__ __forceinline__ float wave_sum32(float v) {
#pragma unroll
  for (int off = 16; off > 0; off >>= 1)
    v += __shfl_xor(v, off, 32);
  return v;
}

__device__ __forceinline__ float softplusf(float x) {
  return (x > 20.0f) ? x : log1pf(expf(x));
}

// ---------------------------------------------------------------------------
// Kernel 1: per-group statistics -> rep[G][D] (written to d_out base).
// One wave32 per group; each lane owns 4 dims (v4f). x streamed non-temporally
// (268 MB, read once, larger than L2) -> pure HBM-bandwidth kernel.
// ---------------------------------------------------------------------------
__global__ __launch_bounds__(256) void group_stats_kernel(
    const float* __restrict__ x, const float* __restrict__ y,
    const float* __restrict__ ax, const float* __restrict__ ay,
    float* __restrict__ rep) {
  const int lane = threadIdx.x & 31;
  const int wave = threadIdx.x >> 5;
  const int g = blockIdx.x * 8 + wave;

  const v4f anc = *(const v4f*)(ax + (size_t)g * D + lane * 4);
  const float ancy = ay[g];

  const float* xb = x + (size_t)g * S * D + lane * 4;
  const float* yb = y + (size_t)g * S;

  v4f sx = {}, sxy = {};
  float sxx = 0.0f, sy = 0.0f;

#pragma unroll 4
  for (int s = 0; s < S; ++s) {
    v4f xv = __builtin_nontemporal_load((const v4f*)(xb + (size_t)s * D));
    __builtin_prefetch(xb + (size_t)(s + 16) * D, 0, 0);  // speculative stream-ahead
    float yr = yb[s] - ancy;
    v4f xr = xv - anc;
    sx += xr;
    sxy += xr * yr;
    sxx += xr.x * xr.x + xr.y * xr.y + xr.z * xr.z + xr.w * xr.w;
    sy += yr;  // identical in all lanes (broadcast y load)
  }

  const float tot  = wave_sum32(sx.x + sx.y + sx.z + sx.w);
  const float sxxT = wave_sum32(sxx);

  const float n  = (float)S;
  const float nD = (float)(S * D);
  v4f cov = (sxy - sx * (sy / n)) * (1.0f / (n - 1.0f));
  float var = (sxxT - tot * tot / nD) / (nD - 1.0f);
  v4f r = cov * (1.0f / var);
  *(v4f*)(rep + (size_t)g * D + lane * 4) = r;
}

// ---------------------------------------------------------------------------
// Kernel 2: MLP head with fp32 WMMA (V_WMMA_F32_16X16X4_F32).
// Block = 256 threads (8 waves), handles 16 rows (groups).
//   layer 1: 8 waves x 2 tiles cover H=256 cols; K-loop 32 x (K=4)
//   layer 2: wave 0, one padded 16x16 tile over K=256
// A fragment layout (16x4 f32): lanes 0-15 -> M=lane, K={k,k+1};
//                               lanes 16-31 -> M=lane-16, K={k+2,k+3}.
// B fragment mirrors per-VGPR: v0={K=k | K=k+2}, v1={K=k+1 | K=k+3}, N=lane&15.
// ---------------------------------------------------------------------------
__global__ __launch_bounds__(256) void mlp_head_kernel(
    const float* __restrict__ rep,
    const float* __restrict__ W1, const float* __restrict__ b1,
    const float* __restrict__ W2, const float* __restrict__ b2,
    float* __restrict__ out) {
  __shared__ float repS[16][132];  // stride 132 -> 4-bank skew, conflict-free
  __shared__ float hS[16][260];
  __shared__ float outS[16][17];

  const int tid  = threadIdx.x;
  const int lane = tid & 31;
  const int wave = tid >> 5;
  const int row0 = blockIdx.x * 16;
  const int nl = lane & 15;
  const int hi = lane >> 4;  // 0: K sub 0/1, 1: K sub 2/3

  // stage rep tile [16][128] into LDS
  for (int t = tid; t < 16 * D; t += 256)
    repS[t >> 7][t & 127] = rep[(size_t)(row0 + (t >> 7)) * D + (t & 127)];
  __syncthreads();

  // ---- layer 1: h = tanh(rep @ W1 + b1) ----
  for (int tile = 0; tile < 2; ++tile) {
    const int n0 = wave * 32 + tile * 16;
    v8f acc = {};
#pragma unroll 8
    for (int k = 0; k < D; k += 4) {
      v2f a, b;
      a.x = repS[nl][k + 2 * hi];
      a.y = repS[nl][k + 2 * hi + 1];
      b.x = W1[(size_t)(k + 2 * hi) * H + n0 + nl];
      b.y = W1[(size_t)(k + 2 * hi + 1) * H + n0 + nl];
      acc = wmma_f32_16x16x4(a, b, acc);
    }
    const float bv = b1[n0 + nl];
#pragma unroll
    for (int j = 0; j < 8; ++j) {
      const int r = hi ? (j + 8) : j;  // C/D layout: lanes 16-31 hold M=j+8
      hS[r][n0 + nl] = tanhf(acc[j] + bv);
    }
  }
  __syncthreads();

  // ---- layer 2: out = h @ W2 + b2 (cols 0..9 valid, padded to 16) ----
  if (wave == 0) {
    v8f acc = {};
#pragma unroll 4
    for (int k = 0; k < H; k += 4) {
      v2f a, b;
      a.x = hS[nl][k + 2 * hi];
      a.y = hS[nl][k + 2 * hi + 1];
      b.x = (nl < KC + 2) ? W2[(size_t)(k + 2 * hi) * (KC + 2) + nl] : 0.0f;
      b.y = (nl < KC + 2) ? W2[(size_t)(k + 2 * hi + 1) * (KC + 2) + nl] : 0.0f;
      acc = wmma_f32_16x16x4(a, b, acc);
    }
    const float bv = (nl < KC + 2) ? b2[nl] : 0.0f;
#pragma unroll
    for (int j = 0; j < 8; ++j) {
      const int r = hi ? (j + 8) : j;
      outS[r][nl] = acc[j] + bv;
    }
  }
  __syncthreads();

  // ---- heads: softplus / softmax / scale, one thread per row ----
  if (tid < 16) {
    const int g = row0 + tid;
    const float alpha = softplusf(outS[tid][0]) * (1.0f - EPS) + EPS;
    const float beta  = softplusf(outS[tid][1]) * (1.0f - EPS) + EPS;

    float m = -__builtin_inff();
#pragma unroll
    for (int i = 0; i < KC; ++i) m = fmaxf(m, outS[tid][2 + i]);
    float e[KC];
    float sum = 0.0f;
#pragma unroll
    for (int i = 0; i < KC; ++i) {
      e[i] = expf(outS[tid][2 + i] - m);
      sum += e[i];
    }
    const float inv = 1.0f / sum;

    float* mix    = out + (size_t)G * D;
    float* scaleP = mix + (size_t)G * KC;
    float* alphaP = scaleP + G;
    float* betaP  = alphaP + G;
#pragma unroll
    for (int i = 0; i < KC; ++i) mix[(size_t)g * KC + i] = e[i] * inv;
    scaleP[g] = sqrtf(beta / alpha);
    alphaP[g] = alpha;
    betaP[g]  = beta;
  }
}

extern "C" void kernel_launch(void* const* d_in, const int* in_sizes, int n_in,
                              void* d_out, int out_size, void* d_ws, size_t ws_size,
                              hipStream_t stream) {
  // setup_inputs order: index, x, y, anchor_x, anchor_y, W1, b1, W2, b2
  const float* x  = (const float*)d_in[1];
  const float* y  = (const float*)d_in[2];
  const float* ax = (const float*)d_in[3];
  const float* ay = (const float*)d_in[4];
  const float* W1 = (const float*)d_in[5];
  const float* b1 = (const float*)d_in[6];
  const float* W2 = (const float*)d_in[7];
  const float* b2 = (const float*)d_in[8];
  float* out = (float*)d_out;

  // rep is written at d_out[0 .. G*D) by kernel 1, then consumed by kernel 2.
  group_stats_kernel<<<G / 8, 256, 0, stream>>>(x, y, ax, ay, out);
  mlp_head_kernel<<<G / 16, 256, 0, stream>>>(out, W1, b1, W2, b2, out);
}